// N2V_GCN_75866302317082
// MI455X (gfx1250) — compile-verified
//
#include <hip/hip_runtime.h>
#include <hip/hip_bf16.h>

typedef __attribute__((ext_vector_type(2))) float v2f;
typedef __attribute__((ext_vector_type(8))) float v8f;

#define NEG_SLOPE 0.2f

// ---------- helpers ----------
__device__ __forceinline__ unsigned f2ord(float f) {
    unsigned u = __float_as_uint(f);
    return (u & 0x80000000u) ? ~u : (u | 0x80000000u);
}
__device__ __forceinline__ float ord2f(unsigned u) {
    return (u & 0x80000000u) ? __uint_as_float(u & 0x7fffffffu)
                             : __uint_as_float(~u);
}
__device__ __forceinline__ float lrelu(float v) { return v >= 0.f ? v : NEG_SLOPE * v; }
__device__ __forceinline__ float elu(float v)   { return v > 0.f ? v : (expf(v) - 1.f); }

// ---------- zero accumulators ----------
__global__ void zero_kernel(float* p, long long count) {
    long long i = (long long)blockIdx.x * blockDim.x + threadIdx.x;
    long long stride = (long long)gridDim.x * blockDim.x;
    for (; i < count; i += stride) p[i] = 0.f;
}

// ---------- f32 WMMA GEMM: C[M,N] = A[M,K] @ B[K,N] ----------
// One wave owns a 16 x (NT*16) strip: the A fragment (b64 load) is reused for
// NT V_WMMA_F32_16X16X4_F32 ops per k-step.
template <int NT>
__global__ void gemm_f32_wmma(const float* __restrict__ A, const float* __restrict__ B,
                              float* __restrict__ C, int M, int N, int K) {
    int wave = (int)((blockIdx.x * (long long)blockDim.x + threadIdx.x) >> 5);
    int lane = threadIdx.x & 31;
    int groups_n = (N >> 4) / NT;               // N is a multiple of NT*16
    int tiles_m  = (M + 15) >> 4;
    if (wave >= tiles_m * groups_n) return;     // wave-uniform: EXEC stays all-1
    int tm = wave / groups_n;
    int tg = wave - tm * groups_n;

    int half = lane >> 4;                       // 0: lanes 0-15, 1: lanes 16-31
    int idx  = lane & 15;
    int row  = tm * 16 + idx;
    int arow = row < M ? row : (M - 1);         // clamp; garbage only feeds unwritten rows
    int koff = half << 1;                       // 0 or 2
    int col0 = tg * (NT * 16) + idx;

    const float* Ar = A + (size_t)arow * K;
    v8f acc[NT];
#pragma unroll
    for (int t = 0; t < NT; ++t) acc[t] = (v8f){};

    for (int k = 0; k < K; k += 4) {
        // A: two consecutive K elements, 8B-aligned -> single b64 load
        v2f a = *(const v2f*)(Ar + k + koff);
        const float* B0 = B + (size_t)(k + koff) * N;
        const float* B1 = B + (size_t)(k + koff + 1) * N;
#pragma unroll
        for (int t = 0; t < NT; ++t) {
            v2f b;
            b.x = B0[col0 + t * 16];
            b.y = B1[col0 + t * 16];
            acc[t] = __builtin_amdgcn_wmma_f32_16x16x4_f32(false, a, false, b,
                                                           (short)0, acc[t], false, false);
        }
    }
    int mbase = tm * 16 + half * 8;             // D: VGPR v -> rows v / v+8 per lane half
#pragma unroll
    for (int t = 0; t < NT; ++t) {
#pragma unroll
        for (int v = 0; v < 8; ++v) {
            int r = mbase + v;
            if (r < M) C[(size_t)r * N + col0 + t * 16] = acc[t][v];
        }
    }
}

// ---------- layer-1 edge pass A: scores + segment max (8 heads, 32 ch) ----------
__global__ void edge1_score(const float* __restrict__ xl, const float* __restrict__ xr,
                            const int* __restrict__ ei, const float* __restrict__ att1,
                            float* __restrict__ score1, unsigned* __restrict__ max1,
                            int E, int n) {
    int wid  = (int)((blockIdx.x * (long long)blockDim.x + threadIdx.x) >> 5);
    int lane = threadIdx.x & 31;
    if (wid >= E + n) return;
    int src = (wid < E) ? ei[wid]     : (wid - E);
    int dst = (wid < E) ? ei[E + wid] : (wid - E);
    const float* pl = xl + (size_t)src * 256;
    const float* pr = xr + (size_t)dst * 256;
#pragma unroll
    for (int h = 0; h < 8; ++h) {
        float t = lrelu(pl[h * 32 + lane] + pr[h * 32 + lane]) * att1[h * 32 + lane];
        for (int off = 16; off; off >>= 1) t += __shfl_xor(t, off, 32);
        if (lane == 0) {
            score1[(size_t)wid * 8 + h] = t;
            atomicMax(&max1[(size_t)dst * 8 + h], f2ord(t));
        }
    }
}

// ---------- layer-1 edge pass B: exp + weighted scatter ----------
__global__ void edge1_agg(const float* __restrict__ xl, const int* __restrict__ ei,
                          const float* __restrict__ score1, const unsigned* __restrict__ max1,
                          float* __restrict__ denom1, float* __restrict__ agg1,
                          int E, int n) {
    int wid  = (int)((blockIdx.x * (long long)blockDim.x + threadIdx.x) >> 5);
    int lane = threadIdx.x & 31;
    if (wid >= E + n) return;
    int src = (wid < E) ? ei[wid]     : (wid - E);
    int dst = (wid < E) ? ei[E + wid] : (wid - E);
    const float* pl = xl + (size_t)src * 256;
#pragma unroll
    for (int h = 0; h < 8; ++h) {
        float m = ord2f(max1[(size_t)dst * 8 + h]);
        float p = expf(score1[(size_t)wid * 8 + h] - m);
        atomicAdd(&agg1[(size_t)dst * 256 + h * 32 + lane], p * pl[h * 32 + lane]);
        if (lane == 0) atomicAdd(&denom1[(size_t)dst * 8 + h], p);
    }
}

// ---------- layer-1 node epilogue: normalize + bias + ELU (in place) ----------
__global__ void node1_elu(float* __restrict__ agg1, const float* __restrict__ denom1,
                          const float* __restrict__ b1, int n) {
    long long tid = (long long)blockIdx.x * blockDim.x + threadIdx.x;
    if (tid >= (long long)n * 256) return;
    int i = (int)(tid >> 8);
    int f = (int)(tid & 255);
    float v = agg1[tid] / denom1[(size_t)i * 8 + (f >> 5)] + b1[f];
    agg1[tid] = elu(v);
}

// ---------- layer-2 edge pass A (1 head, 32 ch) ----------
__global__ void edge2_score(const float* __restrict__ yl, const float* __restrict__ yr,
                            const int* __restrict__ ei, const float* __restrict__ att2,
                            float* __restrict__ score2, unsigned* __restrict__ max2,
                            int E, int n) {
    int wid  = (int)((blockIdx.x * (long long)blockDim.x + threadIdx.x) >> 5);
    int lane = threadIdx.x & 31;
    if (wid >= E + n) return;
    int src = (wid < E) ? ei[wid]     : (wid - E);
    int dst = (wid < E) ? ei[E + wid] : (wid - E);
    float t = lrelu(yl[(size_t)src * 32 + lane] + yr[(size_t)dst * 32 + lane]) * att2[lane];
    for (int off = 16; off; off >>= 1) t += __shfl_xor(t, off, 32);
    if (lane == 0) {
        score2[wid] = t;
        atomicMax(&max2[dst], f2ord(t));
    }
}

// ---------- layer-2 edge pass B ----------
__global__ void edge2_agg(const float* __restrict__ yl, const int* __restrict__ ei,
                          const float* __restrict__ score2, const unsigned* __restrict__ max2,
                          float* __restrict__ denom2, float* __restrict__ agg2,
                          int E, int n) {
    int wid  = (int)((blockIdx.x * (long long)blockDim.x + threadIdx.x) >> 5);
    int lane = threadIdx.x & 31;
    if (wid >= E + n) return;
    int src = (wid < E) ? ei[wid]     : (wid - E);
    int dst = (wid < E) ? ei[E + wid] : (wid - E);
    float p = expf(score2[wid] - ord2f(max2[dst]));
    atomicAdd(&agg2[(size_t)dst * 32 + lane], p * yl[(size_t)src * 32 + lane]);
    if (lane == 0) atomicAdd(&denom2[dst], p);
}

// ---------- layer-2 node epilogue ----------
__global__ void node2_elu(float* __restrict__ agg2, const float* __restrict__ denom2,
                          const float* __restrict__ b2, int n) {
    long long tid = (long long)blockIdx.x * blockDim.x + threadIdx.x;
    if (tid >= (long long)n * 32) return;
    int i = (int)(tid >> 5);
    int c = (int)(tid & 31);
    float v = agg2[tid] / denom2[i] + b2[c];
    agg2[tid] = elu(v);
}

// ---------- classifier: out = [h2, x] @ Wc + bc ----------
__global__ void classifier(const float* __restrict__ h2, const float* __restrict__ x,
                           const float* __restrict__ Wc, const float* __restrict__ bc,
                           float* __restrict__ out, int n) {
    long long tid = (long long)blockIdx.x * blockDim.x + threadIdx.x;
    if (tid >= (long long)n * 40) return;
    int i = (int)(tid / 40);
    int o = (int)(tid - (long long)i * 40);
    const float* h  = h2 + (size_t)i * 32;
    const float* xx = x  + (size_t)i * 128;
    float acc = bc[o];
#pragma unroll 4
    for (int k = 0; k < 32; ++k)  acc += h[k]  * Wc[(size_t)k * 40 + o];
#pragma unroll 4
    for (int k = 0; k < 128; ++k) acc += xx[k] * Wc[(size_t)(32 + k) * 40 + o];
    out[tid] = acc;
}

extern "C" void kernel_launch(void* const* d_in, const int* in_sizes, int n_in,
                              void* d_out, int out_size, void* d_ws, size_t ws_size,
                              hipStream_t stream) {
    const float* x    = (const float*)d_in[0];
    const int*   ei   = (const int*)d_in[1];
    const float* W1l  = (const float*)d_in[2];
    const float* W1r  = (const float*)d_in[3];
    const float* att1 = (const float*)d_in[4];
    const float* b1   = (const float*)d_in[5];
    const float* W2l  = (const float*)d_in[6];
    const float* W2r  = (const float*)d_in[7];
    const float* att2 = (const float*)d_in[8];
    const float* b2   = (const float*)d_in[9];
    const float* Wc   = (const float*)d_in[10];
    const float* bc   = (const float*)d_in[11];
    float* out = (float*)d_out;

    const int n  = in_sizes[0] / 128;   // 50000 nodes
    const int E  = in_sizes[1] / 2;     // 800000 edges
    const int Et = E + n;               // + self-loops

    // ---- workspace layout (floats) ----
    float* ws = (float*)d_ws;
    float* xl = ws;                       ws += (size_t)n * 256;
    float* xr = ws;                       ws += (size_t)n * 256;
    float* zbase  = ws;                   // start of zero-init accumulator region
    float* agg1   = ws;                   ws += (size_t)n * 256;  // becomes h1
    unsigned* max1 = (unsigned*)ws;       ws += (size_t)n * 8;
    float* denom1 = ws;                   ws += (size_t)n * 8;
    float* agg2   = ws;                   ws += (size_t)n * 32;   // becomes h2
    unsigned* max2 = (unsigned*)ws;       ws += (size_t)n;
    float* denom2 = ws;                   ws += (size_t)n;
    long long zcount = (long long)(ws - zbase);
    float* yl     = ws;                   ws += (size_t)n * 32;
    float* yr     = ws;                   ws += (size_t)n * 32;
    float* score1 = ws;                   ws += (size_t)Et * 8;
    float* score2 = ws;                   ws += (size_t)Et;
    size_t need_bytes = (size_t)(ws - (float*)d_ws) * sizeof(float);
    if (ws_size < need_bytes) return;     // deterministic guard

    // 1) zero accumulators (key 0 == ordered -inf for the atomic max buffers)
    zero_kernel<<<2048, 256, 0, stream>>>(zbase, zcount);

    // 2) layer-1 linear transforms via f32 WMMA (16x64 strip per wave)
    {
        int waves = ((n + 15) / 16) * (256 / 64);
        int blocks = (waves + 3) / 4;     // 4 waves / 128-thread block
        gemm_f32_wmma<4><<<blocks, 128, 0, stream>>>(x, W1l, xl, n, 256, 128);
        gemm_f32_wmma<4><<<blocks, 128, 0, stream>>>(x, W1r, xr, n, 256, 128);
    }

    // 3-4) layer-1 attention (wave per edge, lane per channel)
    int eblocks = (Et + 7) / 8;           // 8 waves / 256-thread block
    edge1_score<<<eblocks, 256, 0, stream>>>(xl, xr, ei, att1, score1, max1, E, n);
    edge1_agg  <<<eblocks, 256, 0, stream>>>(xl, ei, score1, max1, denom1, agg1, E, n);

    // 5) normalize + bias + ELU -> h1 (in place in agg1)
    node1_elu<<<(int)(((long long)n * 256 + 255) / 256), 256, 0, stream>>>(agg1, denom1, b1, n);

    // 6) layer-2 linear transforms via f32 WMMA (16x32 strip per wave)
    {
        int waves = ((n + 15) / 16) * (32 / 32);
        int blocks = (waves + 3) / 4;
        gemm_f32_wmma<2><<<blocks, 128, 0, stream>>>(agg1, W2l, yl, n, 32, 256);
        gemm_f32_wmma<2><<<blocks, 128, 0, stream>>>(agg1, W2r, yr, n, 32, 256);
    }

    // 7-8) layer-2 attention
    edge2_score<<<eblocks, 256, 0, stream>>>(yl, yr, ei, att2, score2, max2, E, n);
    edge2_agg  <<<eblocks, 256, 0, stream>>>(yl, ei, score2, max2, denom2, agg2, E, n);

    // 9) normalize + bias + ELU -> h2 (in place in agg2)
    node2_elu<<<(int)(((long long)n * 32 + 255) / 256), 256, 0, stream>>>(agg2, denom2, b2, n);

    // 10) concat classifier
    classifier<<<(int)(((long long)n * 40 + 255) / 256), 256, 0, stream>>>(agg2, x, Wc, bc, out, n);
}